// RGCN_31396210933967
// MI455X (gfx1250) — compile-verified
//
#include <hip/hip_runtime.h>

typedef __attribute__((ext_vector_type(2))) float v2f;
typedef __attribute__((ext_vector_type(8))) float v8f;

#define XS_STRIDE 66  // 16 rows of 64 floats, padded to avoid 16-way LDS bank conflicts

// y[row, basis*BN + n] = x[gather[row], :64] @ V[basis][:, n]   (fp32 WMMA 16x16x4)
// Block = 128 threads = 4 waves; block owns 16 rows; wave w owns basis w.
template <int BN, int NT>
__global__ __launch_bounds__(128) void basis_gemm(
    const float* __restrict__ x,      // [*, 64] source features
    const int*   __restrict__ gather, // nullptr => identity rows
    const float* __restrict__ V,      // [4, 64, BN]
    float*       __restrict__ y)      // [rows, 4*BN]
{
    __shared__ float xs[16 * XS_STRIDE];
    const int tid  = threadIdx.x;
    const int row0 = blockIdx.x * 16;

    // Cooperative gather of the 16x64 fp32 A-tile into LDS (8 floats / thread).
    {
        const int r  = tid >> 3;
        const int c0 = (tid & 7) << 3;
        const int g  = gather ? gather[row0 + r] : (row0 + r);
        const float* src = x + (long)g * 64 + c0;
        float4 p0 = *(const float4*)(src);
        float4 p1 = *(const float4*)(src + 4);
        float* dp = &xs[r * XS_STRIDE + c0];
        dp[0] = p0.x; dp[1] = p0.y; dp[2] = p0.z; dp[3] = p0.w;
        dp[4] = p1.x; dp[5] = p1.y; dp[6] = p1.z; dp[7] = p1.w;
    }
    __syncthreads();

    const int lane  = tid & 31;
    const int basis = tid >> 5;   // wave id 0..3
    const int m     = lane & 15;  // M for A, N for B/D
    const int hi    = lane >> 4;  // lane-half selects K pair / M+8

    const float* __restrict__ Vb = V + basis * (64 * BN);

    v8f acc[NT];
#pragma unroll
    for (int nt = 0; nt < NT; ++nt) acc[nt] = (v8f){0, 0, 0, 0, 0, 0, 0, 0};

#pragma unroll
    for (int k = 0; k < 64; k += 4) {
        const int kk = k + 2 * hi;
        // A 16x4 fp32 fragment: VGPR0 = K=kk, VGPR1 = K=kk+1 (lanes 16-31 get K+2)
        v2f a;
        a.x = xs[m * XS_STRIDE + kk];
        a.y = xs[m * XS_STRIDE + kk + 1];
#pragma unroll
        for (int nt = 0; nt < NT; ++nt) {
            // B 4x16 fp32 fragment (row-striped across lanes, mirroring A)
            v2f b;
            b.x = Vb[kk * BN + nt * 16 + m];
            b.y = Vb[(kk + 1) * BN + nt * 16 + m];
            acc[nt] = __builtin_amdgcn_wmma_f32_16x16x4_f32(
                false, a, false, b, (short)0, acc[nt], false, false);
        }
    }

    // D 16x16 f32 layout: VGPR r -> M=r (lanes 0-15) / M=8+r (lanes 16-31), N=m
#pragma unroll
    for (int nt = 0; nt < NT; ++nt) {
#pragma unroll
        for (int r = 0; r < 8; ++r) {
            const int grow = row0 + r + 8 * hi;
            y[(long)grow * (4 * BN) + basis * BN + nt * 16 + m] = acc[nt][r];
        }
    }
}

// Per-edge: m = norm_e * sum_b comp[etype_e][b] * y[src_e, b, :]; atomic scatter to dst.
template <int OUT>
__global__ __launch_bounds__(256) void edge_scatter(
    const float* __restrict__ y,     // [n_src, 4*OUT]
    const int*   __restrict__ src,
    const int*   __restrict__ dst,
    const int*   __restrict__ etype,
    const float* __restrict__ norm,  // [E]
    const float* __restrict__ comp,  // [64, 4]
    float*       __restrict__ hout,  // [n_dst, OUT]
    int E)
{
    const int EPB = 256 / OUT;
    const int e = blockIdx.x * EPB + threadIdx.x / OUT;
    const int o = threadIdx.x & (OUT - 1);
    if (e >= E) return;
    const int   s = src[e];
    const int   d = dst[e];
    const int   t = etype[e];
    const float nrm = norm[e];
    const float c0 = comp[t * 4 + 0];
    const float c1 = comp[t * 4 + 1];
    const float c2 = comp[t * 4 + 2];
    const float c3 = comp[t * 4 + 3];
    const float* __restrict__ ys = y + (long)s * (4 * OUT);
    const float mval =
        (c0 * ys[o] + c1 * ys[OUT + o] + c2 * ys[2 * OUT + o] + c3 * ys[3 * OUT + o]) * nrm;
    atomicAdd(&hout[(long)d * OUT + o], mval);
}

__global__ __launch_bounds__(256) void bias_relu(float* __restrict__ h,
                                                 const float* __restrict__ b, int n) {
    const int i = blockIdx.x * 256 + threadIdx.x;
    if (i < n) {
        const float v = h[i] + b[i & 63];
        h[i] = v > 0.f ? v : 0.f;
    }
}

__global__ __launch_bounds__(256) void init_bias(float* __restrict__ out,
                                                 const float* __restrict__ b, int n) {
    const int i = blockIdx.x * 256 + threadIdx.x;
    if (i < n) out[i] = b[i & 31];
}

extern "C" void kernel_launch(void* const* d_in, const int* in_sizes, int n_in,
                              void* d_out, int out_size, void* d_ws, size_t ws_size,
                              hipStream_t stream) {
    const int*   input_nodes = (const int*)d_in[0];
    const int*   src1        = (const int*)d_in[1];
    const int*   dst1        = (const int*)d_in[2];
    const int*   etype1      = (const int*)d_in[3];
    const float* norm1       = (const float*)d_in[4];
    const int*   src2        = (const int*)d_in[5];
    const int*   dst2        = (const int*)d_in[6];
    const int*   etype2      = (const int*)d_in[7];
    const float* norm2       = (const float*)d_in[8];
    const float* emb         = (const float*)d_in[9];
    const float* V1          = (const float*)d_in[10];
    const float* comp1       = (const float*)d_in[11];
    const float* b1          = (const float*)d_in[12];
    const float* V2          = (const float*)d_in[13];
    const float* comp2       = (const float*)d_in[14];
    const float* b2          = (const float*)d_in[15];
    float*       out         = (float*)d_out;

    const int N_SRC0 = 400000, N_DST1 = 100000, N_DST2 = 20000;
    const int E1 = 600000, E2 = 300000;

    // Workspace layout (all fp32): y1 [400000,256] | h [100000,64] | y2 [100000,128]
    float* y1 = (float*)d_ws;
    float* h  = y1 + (size_t)N_SRC0 * 256;
    float* y2 = h + (size_t)N_DST1 * 64;

    hipMemsetAsync(h, 0, (size_t)N_DST1 * 64 * sizeof(float), stream);

    // Layer 1: gather emb rows + 4-basis GEMM (K=64, per-basis N=64)
    basis_gemm<64, 4><<<N_SRC0 / 16, 128, 0, stream>>>(emb, input_nodes, V1, y1);
    edge_scatter<64><<<E1 / 4, 256, 0, stream>>>(y1, src1, dst1, etype1, norm1, comp1, h, E1);
    bias_relu<<<(N_DST1 * 64 + 255) / 256, 256, 0, stream>>>(h, b1, N_DST1 * 64);

    // Layer 2: no gather, per-basis N=32
    basis_gemm<32, 2><<<N_DST1 / 16, 128, 0, stream>>>(h, nullptr, V2, y2);
    init_bias<<<(N_DST2 * 32 + 255) / 256, 256, 0, stream>>>(out, b2, N_DST2 * 32);
    edge_scatter<32><<<E2 / 8, 256, 0, stream>>>(y2, src2, dst2, etype2, norm2, comp2, out, E2);
}